// SelfAttention_63737314673092
// MI455X (gfx1250) — compile-verified
//
#include <hip/hip_runtime.h>
#include <math.h>

#define Bz 16
#define Sx 512
#define Dx 512
#define Hx 8
#define DHx 64
#define SDx (Sx * Dx)      // 262144
#define RETx 512
#define LNEPS 1e-12f

#define KCHUNK 2048
#define NKB (SDx / KCHUNK) // 128
#define TK 64

typedef float v2f __attribute__((ext_vector_type(2)));
typedef float v8f __attribute__((ext_vector_type(8)));

// fp32 WMMA: D(16x16,f32) = A(16x4,f32) * B(4x16,f32) + C
__device__ __forceinline__ v8f wmma4(v2f a, v2f b, v8f c) {
  return __builtin_amdgcn_wmma_f32_16x16x4_f32(
      /*neg_a=*/false, a, /*neg_b=*/false, b,
      /*c_mod=*/(short)0, c, /*reuse_a=*/false, /*reuse_b=*/false);
}

__device__ __forceinline__ float swishf(float v) {
  return v / (1.0f + __expf(-v));
}

// ---------------------------------------------------------------------------
// Stage 1: Q/K/V projections.  C[8192x512] = x[8192x512] * W[512x512]
// block = 256 threads (8 waves); wave w -> 16x64 tile (n0 = w*64)
// grid = (8192/16, 3)  (z-index in grid.y selects WQ/WK/WV)
// ---------------------------------------------------------------------------
__global__ void qkv_proj(const float* __restrict__ x,
                         const float* __restrict__ WQ,
                         const float* __restrict__ WK,
                         const float* __restrict__ WV,
                         float* __restrict__ Q,
                         float* __restrict__ Kp,
                         float* __restrict__ V) {
  const int wz = blockIdx.y;
  const float* W = (wz == 0) ? WQ : (wz == 1) ? WK : WV;
  float* out = (wz == 0) ? Q : (wz == 1) ? Kp : V;

  const int m0   = blockIdx.x * 16;
  const int wave = threadIdx.x >> 5;
  const int lane = threadIdx.x & 31;
  const int r    = lane & 15;          // row within tile (A) / col (B,C)
  const int kk   = (lane >> 4) * 2;    // K sub-offset per lane half
  const int n0   = wave * 64;

  v8f a0 = {}, a1 = {}, a2 = {}, a3 = {};
  const float* pa_base = x + (size_t)(m0 + r) * Dx + kk;

  for (int k0 = 0; k0 < Dx; k0 += 4) {
    v2f af = *(const v2f*)(pa_base + k0);
    const float* pb = W + (size_t)(k0 + kk) * Dx + n0 + r;
    v2f b0 = { pb[0],       pb[Dx] };
    v2f b1 = { pb[16],      pb[16 + Dx] };
    v2f b2 = { pb[32],      pb[32 + Dx] };
    v2f b3 = { pb[48],      pb[48 + Dx] };
    a0 = wmma4(af, b0, a0);
    a1 = wmma4(af, b1, a1);
    a2 = wmma4(af, b2, a2);
    a3 = wmma4(af, b3, a3);
  }

  const int rb = (lane >> 4) * 8;
  float* o = out + (size_t)(m0 + rb) * Dx + n0 + r;
#pragma unroll
  for (int j = 0; j < 8; ++j) {
    o[(size_t)j * Dx + 0]  = a0[j];
    o[(size_t)j * Dx + 16] = a1[j];
    o[(size_t)j * Dx + 32] = a2[j];
    o[(size_t)j * Dx + 48] = a3[j];
  }
}

// ---------------------------------------------------------------------------
// Stage 2: streaming swish-attention (no softmax -> single pass over keys).
// Each wave: 16 queries of one (b,h); loop keys in chunks of 16:
//   scores(16x16) = Q(16x64) * K^T(64x16) via 16 wmma, swish,
//   LDS round-trip to A-layout, O(16x64) += scores * V via 16 wmma.
// block = 256 (8 waves, 128 queries); grid = (S/128, H, B)
// ---------------------------------------------------------------------------
__global__ void attn_kernel(const float* __restrict__ Q,
                            const float* __restrict__ Kp,
                            const float* __restrict__ V,
                            float* __restrict__ attn) {
  const int b    = blockIdx.z;
  const int h    = blockIdx.y;
  const int wave = threadIdx.x >> 5;
  const int lane = threadIdx.x & 31;
  const int q0   = blockIdx.x * 128 + wave * 16;
  const int r    = lane & 15;
  const int kk   = (lane >> 4) * 2;
  const int rb   = (lane >> 4) * 8;
  const float scale = 0.125f;  // 1/sqrt(64)

  __shared__ float sc[8][16][17];  // per-wave score tile, padded

  // Preload Q A-fragments covering dh=64 (16 K-steps of 4)
  v2f qf[16];
  const float* qp = Q + ((size_t)b * Sx + q0 + r) * Dx + h * DHx + kk;
#pragma unroll
  for (int i = 0; i < 16; ++i) qf[i] = *(const v2f*)(qp + i * 4);

  v8f o0 = {}, o1 = {}, o2 = {}, o3 = {};

  for (int kt = 0; kt < Sx; kt += 16) {
    // scores tile: B-frag is K^T -> element (k,n) = K[key n][dim k]
    v8f c = {};
    const float* kpp = Kp + ((size_t)b * Sx + kt + r) * Dx + h * DHx + kk;
#pragma unroll
    for (int i = 0; i < 16; ++i) {
      v2f bf = *(const v2f*)(kpp + i * 4);
      c = wmma4(qf[i], bf, c);
    }
#pragma unroll
    for (int j = 0; j < 8; ++j) c[j] = swishf(c[j] * scale);

    __syncthreads();
#pragma unroll
    for (int j = 0; j < 8; ++j) sc[wave][rb + j][r] = c[j];
    __syncthreads();

    // O += scores(16x16) * V(16x64): 4 K-steps x 4 N-tiles
    const float* vp = V + ((size_t)b * Sx + kt + kk) * Dx + h * DHx + r;
#pragma unroll
    for (int kc = 0; kc < 4; ++kc) {
      v2f af = { sc[wave][r][kc * 4 + kk], sc[wave][r][kc * 4 + kk + 1] };
      const float* vrow = vp + (size_t)(kc * 4) * Dx;
      v2f b0 = { vrow[0],  vrow[Dx] };
      v2f b1 = { vrow[16], vrow[16 + Dx] };
      v2f b2 = { vrow[32], vrow[32 + Dx] };
      v2f b3 = { vrow[48], vrow[48 + Dx] };
      o0 = wmma4(af, b0, o0);
      o1 = wmma4(af, b1, o1);
      o2 = wmma4(af, b2, o2);
      o3 = wmma4(af, b3, o3);
    }
  }

  float* op = attn + ((size_t)b * Sx + q0 + rb) * Dx + h * DHx + r;
#pragma unroll
  for (int j = 0; j < 8; ++j) {
    op[(size_t)j * Dx + 0]  = o0[j];
    op[(size_t)j * Dx + 16] = o1[j];
    op[(size_t)j * Dx + 32] = o2[j];
    op[(size_t)j * Dx + 48] = o3[j];
  }
}

// ---------------------------------------------------------------------------
// Stage 3: residual add + TF LayerNorm over D=512. One wave per row.
// block = 256 (8 rows); grid = 8192/8
// ---------------------------------------------------------------------------
__global__ void resid_ln(const float* __restrict__ attn,
                         const float* __restrict__ x,
                         const float* __restrict__ lnw,
                         const float* __restrict__ lnb,
                         float* __restrict__ normed) {
  const int wave = threadIdx.x >> 5;
  const int lane = threadIdx.x & 31;
  const int row  = blockIdx.x * 8 + wave;

  const float* a  = attn + (size_t)row * Dx;
  const float* xr = x + (size_t)row * Dx;

  float vals[16];
  float s = 0.f;
#pragma unroll
  for (int i = 0; i < 16; ++i) {
    float v = a[lane + i * 32] + xr[lane + i * 32];
    vals[i] = v;
    s += v;
  }
#pragma unroll
  for (int off = 16; off > 0; off >>= 1) s += __shfl_xor(s, off, 32);
  const float mean = s * (1.0f / Dx);

  float vs = 0.f;
#pragma unroll
  for (int i = 0; i < 16; ++i) {
    float d2 = vals[i] - mean;
    vs += d2 * d2;
  }
#pragma unroll
  for (int off = 16; off > 0; off >>= 1) vs += __shfl_xor(vs, off, 32);
  const float inv = 1.0f / sqrtf(vs * (1.0f / Dx) + LNEPS);

  float* np = normed + (size_t)row * Dx;
#pragma unroll
  for (int i = 0; i < 16; ++i) {
    int col = lane + i * 32;
    np[col] = lnw[col] * ((vals[i] - mean) * inv) + lnb[col];
  }
}

// ---------------------------------------------------------------------------
// Stage 4: split-K final linear. y[16,512] = normed[16,SD] @ Wlin[SD,512].
// 128 blocks x 2048 K; thread t owns column t; Wlin rows read coalesced once.
// ---------------------------------------------------------------------------
__global__ void lin_partial(const float* __restrict__ normed,
                            const float* __restrict__ Wlin,
                            float* __restrict__ partial) {
  __shared__ float sa[16][TK];
  const int n   = threadIdx.x;           // 0..511
  const int kc0 = blockIdx.x * KCHUNK;

  float acc[16];
#pragma unroll
  for (int bb = 0; bb < 16; ++bb) acc[bb] = 0.f;

  for (int kt = 0; kt < KCHUNK; kt += TK) {
    __syncthreads();
    for (int i = threadIdx.x; i < 16 * TK; i += 512) {
      sa[i >> 6][i & (TK - 1)] =
          normed[(size_t)(i >> 6) * SDx + kc0 + kt + (i & (TK - 1))];
    }
    __syncthreads();

    const float* wp = Wlin + (size_t)(kc0 + kt) * RETx + n;
#pragma unroll 4
    for (int k = 0; k < TK; ++k) {
      float w = wp[(size_t)k * RETx];
#pragma unroll
      for (int bb = 0; bb < 16; ++bb) acc[bb] += sa[bb][k] * w;
    }
  }

  float* pp = partial + (size_t)blockIdx.x * (16 * RETx) + n;
#pragma unroll
  for (int bb = 0; bb < 16; ++bb) pp[(size_t)bb * RETx] = acc[bb];
}

// Stage 5: reduce split-K partials, add bias, swish.
__global__ void lin_reduce(const float* __restrict__ partial,
                           const float* __restrict__ blin,
                           float* __restrict__ out) {
  const int i = blockIdx.x * 256 + threadIdx.x;  // 0..8191
  float s = 0.f;
  for (int j = 0; j < NKB; ++j) s += partial[(size_t)j * (16 * RETx) + i];
  s += blin[i & (RETx - 1)];
  out[i] = swishf(s);
}

// ---------------------------------------------------------------------------
extern "C" void kernel_launch(void* const* d_in, const int* in_sizes, int n_in,
                              void* d_out, int out_size, void* d_ws,
                              size_t ws_size, hipStream_t stream) {
  const float* x    = (const float*)d_in[0];
  const float* WQ   = (const float*)d_in[1];
  const float* WK   = (const float*)d_in[2];
  const float* WV   = (const float*)d_in[3];
  const float* Wlin = (const float*)d_in[4];
  const float* blin = (const float*)d_in[5];
  const float* lnw  = (const float*)d_in[6];
  const float* lnb  = (const float*)d_in[7];

  float* ws = (float*)d_ws;
  const size_t NBSD = (size_t)Bz * SDx;  // 4,194,304 floats each
  float* Q      = ws;
  float* Kp     = Q + NBSD;
  float* V      = Kp + NBSD;
  float* attn   = V + NBSD;
  float* normed = attn + NBSD;
  float* part   = normed + NBSD;         // 128*16*512 floats

  qkv_proj<<<dim3((Bz * Sx) / 16, 3), 256, 0, stream>>>(x, WQ, WK, WV, Q, Kp, V);
  attn_kernel<<<dim3(Sx / 128, Hx, Bz), 256, 0, stream>>>(Q, Kp, V, attn);
  resid_ln<<<dim3((Bz * Sx) / 8), 256, 0, stream>>>(attn, x, lnw, lnb, normed);
  lin_partial<<<dim3(NKB), 512, 0, stream>>>(normed, Wlin, part);
  lin_reduce<<<dim3((Bz * RETx) / 256), 256, 0, stream>>>(part, blin, (float*)d_out);
}